// NeuralStack_14070312861728
// MI455X (gfx1250) — compile-verified
//
#include <hip/hip_runtime.h>

typedef __attribute__((ext_vector_type(2))) float v2f;
typedef __attribute__((ext_vector_type(8))) float v8f;

#define T1_   4095           // existing stack entries
#define TT_   4096           // after push
#define B_    64
#define M_    128
#define NCHUNK 16
#define CHUNK  256           // TT_ / NCHUNK

// ---------------------------------------------------------------------------
// Kernel 1: per-batch scans.  One block per batch column b.
// Computes sn (to d_out) and coeff (to workspace) via two forward inclusive
// block scans; reverse-exclusive sums derived as (total - inclusive).
// ---------------------------------------------------------------------------
__global__ __launch_bounds__(256) void ns_scan_kernel(
    const float* __restrict__ s, const float* __restrict__ d,
    const float* __restrict__ u, float* __restrict__ sn_out,
    float* __restrict__ coeff)
{
    __shared__ float lds[256];
    const int b   = blockIdx.x;
    const int tid = threadIdx.x;
    const float ub = u[b];
    const float db = d[b];

    float loc[16], pref[16];
    #pragma unroll
    for (int k = 0; k < 16; ++k) {
        int i = tid * 16 + k;
        loc[k] = (i < T1_) ? s[i * B_ + b] : 0.0f;     // pad slot 4095 with 0
    }
    float run = 0.f;
    #pragma unroll
    for (int k = 0; k < 16; ++k) { run += loc[k]; pref[k] = run; }

    lds[tid] = run;
    __syncthreads();
    #pragma unroll
    for (int off = 1; off < 256; off <<= 1) {
        float t = (tid >= off) ? lds[tid - off] : 0.f;
        __syncthreads();
        lds[tid] += t;
        __syncthreads();
    }
    float total = lds[255];
    float excl  = lds[tid] - run;

    float snv[16];
    #pragma unroll
    for (int k = 0; k < 16; ++k) {
        int i = tid * 16 + k;
        float incl    = excl + pref[k];
        float totprev = total - incl;                   // sum_{j>i} s[j]
        float t1 = ub - totprev;  t1 = t1 > 0.f ? t1 : 0.f;
        float t2 = loc[k] - t1;   t2 = t2 > 0.f ? t2 : 0.f;
        float snval = (i < T1_) ? t2 : db;              // sn[T-1] = d
        snv[k] = snval;
        sn_out[i * B_ + b] = snval;
    }
    __syncthreads();

    // second scan over sn (full length TT_)
    run = 0.f;
    #pragma unroll
    for (int k = 0; k < 16; ++k) { run += snv[k]; pref[k] = run; }
    lds[tid] = run;
    __syncthreads();
    #pragma unroll
    for (int off = 1; off < 256; off <<= 1) {
        float t = (tid >= off) ? lds[tid - off] : 0.f;
        __syncthreads();
        lds[tid] += t;
        __syncthreads();
    }
    total = lds[255];
    excl  = lds[tid] - run;

    #pragma unroll
    for (int k = 0; k < 16; ++k) {
        int i = tid * 16 + k;
        float incl = excl + pref[k];
        float tots = total - incl;                      // sum_{j>i} sn[j]
        float rem  = 1.0f - tots; rem = rem > 0.f ? rem : 0.f;
        float c    = snv[k] < rem ? snv[k] : rem;       // min(sn, relu(1-tots))
        coeff[b * TT_ + i] = c;
    }
}

// ---------------------------------------------------------------------------
// Kernel 2: fused copy (V,v -> Vn) + WMMA reduction partials.
// grid = (NCHUNK t-chunks, B batches), 256 threads = 8 waves, one 16-wide
// m-tile per wave.  Main loop is fully branch-free (no EXEC manipulation):
// unconditional NT load of V rows in the v_wmma_f32_16x16x4_f32 B layout,
// NT store to Vn, coeff pair load, wmma accumulate.  The push-row boundary
// (t == 4095) exists only in the last 4-row group of the last chunk; it is
// handled by a block-uniform epilogue using a clamped in-bounds load plus a
// lane select (v_cndmask), keeping EXEC all-ones around the WMMA.
// ---------------------------------------------------------------------------
__global__ __launch_bounds__(256) void ns_stream_kernel(
    const float* __restrict__ V, const float* __restrict__ v,
    const float* __restrict__ coeff, float* __restrict__ Vn,
    float* __restrict__ part)
{
    const int b     = blockIdx.y;
    const int chunk = blockIdx.x;
    const int t0    = chunk * CHUNK;
    const int wave  = threadIdx.x >> 5;
    const int lane  = threadIdx.x & 31;
    const int m0    = wave * 16;
    const int n     = lane & 15;
    const int khalf = (lane >> 4) * 2;   // 0 for lanes 0-15, 2 for lanes 16-31

    const int   colbase = b * M_ + m0 + n;   // also the index into v[b, m0+n]
    const float* cb     = coeff + b * TT_;

    v8f acc = {};
    const bool last  = (chunk == NCHUNK - 1);
    const int  nfull = last ? (CHUNK - 4) : CHUNK;   // block-uniform

    int r0 = t0 + khalf;
    for (int tt = 0; tt < nfull; tt += 4) {
        int i0 = r0 * (B_ * M_) + colbase;
        int i1 = i0 + (B_ * M_);

        float val0 = __builtin_nontemporal_load(V + i0);
        float val1 = __builtin_nontemporal_load(V + i1);
        __builtin_nontemporal_store(val0, Vn + i0);
        __builtin_nontemporal_store(val1, Vn + i1);

        v2f a;  a[0]  = cb[r0]; a[1]  = cb[r0 + 1];
        v2f bf; bf[0] = val0;   bf[1] = val1;
        acc = __builtin_amdgcn_wmma_f32_16x16x4_f32(
            false, a, false, bf, (short)0, acc, false, false);

        r0 += 4;
    }

    if (last) {
        // rows 4092..4095 ; r0 in {4092,4094}, r1 in {4093,4095}
        int r1 = r0 + 1;
        float vlast = v[colbase];
        int i0  = r0 * (B_ * M_) + colbase;
        int r1c = (r1 < T1_) ? r1 : (T1_ - 1);           // clamp: stay in-bounds
        int i1c = r1c * (B_ * M_) + colbase;
        int i1  = r1  * (B_ * M_) + colbase;

        float val0  = __builtin_nontemporal_load(V + i0);
        float vload = __builtin_nontemporal_load(V + i1c);
        float val1  = (r1 < T1_) ? vload : vlast;        // select, not branch
        __builtin_nontemporal_store(val0, Vn + i0);
        __builtin_nontemporal_store(val1, Vn + i1);

        v2f a;  a[0]  = cb[r0]; a[1]  = cb[r1];
        v2f bf; bf[0] = val0;   bf[1] = val1;
        acc = __builtin_amdgcn_wmma_f32_16x16x4_f32(
            false, a, false, bf, (short)0, acc, false, false);
    }

    if (lane < 16)
        part[(chunk * B_ + b) * M_ + m0 + lane] = acc[0];
}

// ---------------------------------------------------------------------------
// Kernel 3: sum the NCHUNK partials into r (plain stores -> deterministic,
// no dependence on poisoned d_out contents).
// ---------------------------------------------------------------------------
__global__ __launch_bounds__(256) void ns_reduce_kernel(
    const float* __restrict__ part, float* __restrict__ r)
{
    int i = blockIdx.x * blockDim.x + threadIdx.x;   // 0 .. B_*M_-1
    float acc = 0.f;
    #pragma unroll
    for (int p = 0; p < NCHUNK; ++p) acc += part[p * (B_ * M_) + i];
    r[i] = acc;
}

extern "C" void kernel_launch(void* const* d_in, const int* in_sizes, int n_in,
                              void* d_out, int out_size, void* d_ws, size_t ws_size,
                              hipStream_t stream)
{
    const float* V = (const float*)d_in[0];   // [4095,64,128]
    const float* s = (const float*)d_in[1];   // [4095,64,1]
    const float* d = (const float*)d_in[2];   // [64,1]
    const float* u = (const float*)d_in[3];   // [64,1]
    const float* v = (const float*)d_in[4];   // [64,128]

    float* out = (float*)d_out;
    float* Vn  = out;                                   // [4096,64,128]
    float* sn  = out + (size_t)TT_ * B_ * M_;           // [4096,64]
    float* r   = sn  + (size_t)TT_ * B_;                // [64,128]

    float* coeff = (float*)d_ws;                        // [64,4096]
    float* part  = coeff + (size_t)B_ * TT_;            // [16,64,128]

    ns_scan_kernel<<<B_, 256, 0, stream>>>(s, d, u, sn, coeff);

    dim3 grid(NCHUNK, B_);
    ns_stream_kernel<<<grid, 256, 0, stream>>>(V, v, coeff, Vn, part);

    ns_reduce_kernel<<<(B_ * M_) / 256, 256, 0, stream>>>(part, r);
}